// SymmetryAwareLossLoop_63574105916003
// MI455X (gfx1250) — compile-verified
//
#include <hip/hip_runtime.h>
#include <hip/hip_bf16.h>
#include <math.h>

typedef __attribute__((ext_vector_type(2))) float v2f;
typedef __attribute__((ext_vector_type(8))) float v8f;

#define GEO_EPS 1e-6f
#define KPAD 12   // K = 9 padded to 12 (3 x WMMA 16x16x4 steps), rows 8B-aligned

// Wave-level max via ds_swizzle (group-of-32 XOR mode): avoids the
// bounds-check/cndmask clamping that __shfl_xor generates.
__device__ __forceinline__ float wave_max32(float x) {
    int v;
    v = __builtin_amdgcn_ds_swizzle(__float_as_int(x), (16 << 10) | 0x1f);
    x = fmaxf(x, __int_as_float(v));
    v = __builtin_amdgcn_ds_swizzle(__float_as_int(x), (8 << 10) | 0x1f);
    x = fmaxf(x, __int_as_float(v));
    v = __builtin_amdgcn_ds_swizzle(__float_as_int(x), (4 << 10) | 0x1f);
    x = fmaxf(x, __int_as_float(v));
    v = __builtin_amdgcn_ds_swizzle(__float_as_int(x), (2 << 10) | 0x1f);
    x = fmaxf(x, __int_as_float(v));
    v = __builtin_amdgcn_ds_swizzle(__float_as_int(x), (1 << 10) | 0x1f);
    x = fmaxf(x, __int_as_float(v));
    return x;
}

// One block per batch element b. 128 threads = 4 waves (wave32).
// Wave w computes the 16x16 trace tile for candidates [w*16, w*16+16) x all
// symmetry ops via V_WMMA_F32_16X16X4_F32 (K = 9 padded to 12 -> 3 WMMA steps).
__global__ void __launch_bounds__(128)
geo_loss_kernel(const float* __restrict__ R_pred,
                const float* __restrict__ R_gt,
                const float* __restrict__ rot_mats,
                float* __restrict__ loss_b,
                int C, int S)
{
    const int b    = blockIdx.x;
    const int tid  = threadIdx.x;
    const int lane = tid & 31;
    const int wave = tid >> 5;

    __shared__ __align__(16) float lds_sym[16 * KPAD];   // R_gt_sym[s][k], zero-padded
    __shared__ __align__(16) float lds_pred[64 * KPAD];  // 64 candidates, padded rows
    __shared__ float red[4];

    // ---- Zero pass: sym matrix (full) + pred pad cols (k = 9..11), only first 128 of 192 ----
    for (int t = tid; t < 16 * KPAD; t += 128) lds_sym[t] = 0.0f;
    if (tid < 64 * 3) lds_pred[(tid / 3) * KPAD + 9 + (tid % 3)] = 0.0f;
    __syncthreads();

    // ---- R_gt_sym[s] = rot_mats[s] @ R_gt[b], flattened k = i*3+j ----
    if (tid < S * 9) {
        const int s = tid / 9, e = tid % 9, i = e / 3, j = e % 3;
        const float* Rm = rot_mats + s * 9;
        const float* Rg = R_gt + (size_t)b * 9;
        lds_sym[s * KPAD + e] =
            Rm[i * 3 + 0] * Rg[0 * 3 + j] +
            Rm[i * 3 + 1] * Rg[1 * 3 + j] +
            Rm[i * 3 + 2] * Rg[2 * 3 + j];
    }

    // A-fragment addressing per ISA 7.12.2 (32-bit A 16x4):
    //   lanes 0-15: M=lane, VGPR v holds K = 2*(lane/16) + v
    const int m     = lane & 15;   // candidate row within tile
    const int khalf = lane >> 4;   // K-half within each 4-chunk
    const int n     = m;           // B/D column (symmetry op) held by this lane
    const int rowB  = n * KPAD;
    const bool nvalid = (n < S);

    float vmax = -INFINITY;

    for (int cbase = 0; cbase < C; cbase += 64) {
        __syncthreads();
        // ---- Coalesced b128 load of 64 candidates (2304 B) into padded LDS ----
        const float4* src4 =
            (const float4*)(R_pred + ((size_t)b * C + cbase) * 9);
        for (int t = tid; t < (64 * 9) / 4; t += 128) {
            const float4 v = src4[t];
            const int g = t * 4;
            lds_pred[((g + 0) / 9) * KPAD + (g + 0) % 9] = v.x;
            lds_pred[((g + 1) / 9) * KPAD + (g + 1) % 9] = v.y;
            lds_pred[((g + 2) / 9) * KPAD + (g + 2) % 9] = v.z;
            lds_pred[((g + 3) / 9) * KPAD + (g + 3) % 9] = v.w;
        }
        __syncthreads();

        // ---- 16x16 trace tile = A(16x12) x B(12x16), 3 WMMA steps ----
        const int rowA = (wave * 16 + m) * KPAD;
        v8f acc = {};
#pragma unroll
        for (int kk = 0; kk < 3; ++kk) {
            const int k0 = kk * 4 + khalf * 2;          // even -> 8B aligned
            const v2f a  = *(const v2f*)&lds_pred[rowA + k0];
            const v2f bm = *(const v2f*)&lds_sym[rowB + k0];
            acc = __builtin_amdgcn_wmma_f32_16x16x4_f32(
                /*neg_a=*/false, a, /*neg_b=*/false, bm,
                /*c_mod=*/(short)0, acc, /*reuse_a=*/false, /*reuse_b=*/false);
        }

        // ---- Masked max over the tile (only real symmetry columns) ----
        float tmax = fmaxf(fmaxf(fmaxf(acc[0], acc[1]), fmaxf(acc[2], acc[3])),
                           fmaxf(fmaxf(acc[4], acc[5]), fmaxf(acc[6], acc[7])));
        vmax = nvalid ? fmaxf(vmax, tmax) : vmax;
    }

    // Wave-level max reduce, then cross-wave via LDS.
    vmax = wave_max32(vmax);
    if (lane == 0) red[wave] = vmax;
    __syncthreads();

    if (tid == 0) {
        // min over (c,s) of arccos(clip(.)) == arccos(clip(max tr)) by monotonicity
        const float mv = fmaxf(fmaxf(red[0], red[1]), fmaxf(red[2], red[3]));
        float c = (mv - 1.0f) * 0.5f;
        c = fminf(fmaxf(c, -1.0f + GEO_EPS), 1.0f - GEO_EPS);
        loss_b[b] = acosf(c);
    }
}

// Deterministic single-block mean over per-b losses (no fp atomics).
__global__ void __launch_bounds__(256)
mean_kernel(const float* __restrict__ loss_b, float* __restrict__ out, int B)
{
    __shared__ float sh[256];
    float s = 0.0f;
    for (int i = threadIdx.x; i < B; i += 256) s += loss_b[i];
    sh[threadIdx.x] = s;
    __syncthreads();
#pragma unroll
    for (int off = 128; off > 0; off >>= 1) {
        if ((int)threadIdx.x < off) sh[threadIdx.x] += sh[threadIdx.x + off];
        __syncthreads();
    }
    if (threadIdx.x == 0) out[0] = sh[0] / (float)B;
}

extern "C" void kernel_launch(void* const* d_in, const int* in_sizes, int n_in,
                              void* d_out, int out_size, void* d_ws, size_t ws_size,
                              hipStream_t stream)
{
    const float* R_pred   = (const float*)d_in[0];  // (B, C, 3, 3) fp32
    const float* R_gt     = (const float*)d_in[1];  // (B, 3, 3)    fp32
    const float* rot_mats = (const float*)d_in[2];  // (S, 3, 3)    fp32

    const int B = in_sizes[1] / 9;                  // 32768
    const int C = in_sizes[0] / (B * 9);            // 64
    const int S = in_sizes[2] / 9;                  // 12

    float* loss_b = (float*)d_ws;                   // B floats of scratch

    geo_loss_kernel<<<B, 128, 0, stream>>>(R_pred, R_gt, rot_mats, loss_b, C, S);
    mean_kernel<<<1, 256, 0, stream>>>(loss_b, (float*)d_out, B);
}